// Lattice_46239617908982
// MI455X (gfx1250) — compile-verified
//
#include <hip/hip_runtime.h>

// MI455X / gfx1250 — multilinear interpolation on a 3^8 lattice reformulated as
//   out[b] = u_b^T (K2) v_b,  K2 = reshape(kernel, 81x81)
// with the heavy middle contraction Z = V * K2^T done via V_WMMA_F32_16X16X4_F32
// (f32 WMMA keeps exact reference precision). Roofline: ~1.2 MB total I/O at
// 23.3 TB/s -> this is latency bound; WMMA covers the 430 MFLOP GEMM for free.

typedef __attribute__((ext_vector_type(2))) float v2f;
typedef __attribute__((ext_vector_type(4))) float v4f;
typedef __attribute__((ext_vector_type(8))) float v8f;

#define WAVES_PER_BLOCK 2           // 64 threads/block, 2 wave32
#define BLOCKS          1024        // 32768 rows / (16 rows * 2 waves)

__global__ __launch_bounds__(64) void Lattice_46239617908982_kernel(
    const float* __restrict__ x,     // [32768][8]
    const float* __restrict__ kern,  // [6561]
    float* __restrict__ out)         // [32768]
{
    // K2 padded to 96 rows (N tiles of 16) x 84 cols (K blocks of 4), zeros outside 81x81.
    __shared__ __align__(16) float K2s[96][84];
    __shared__ __align__(16) float Vs[WAVES_PER_BLOCK][16][84];  // dense V tile, cols>=81 zero
    __shared__ __align__(16) float Zs[WAVES_PER_BLOCK][16][96];  // Z tile (N padded to 96)
    __shared__             float W1[WAVES_PER_BLOCK][16][12];    // hat weights dims 4..7

    const int tid  = threadIdx.x;
    const int wv   = tid >> 5;
    const int lane = tid & 31;
    const int m16  = lane & 15;      // row within tile / column within B tile
    const int hi   = lane >> 4;      // lane-half -> k-subblock select

    // ---- stage K2 (81x81, row-major: flat = hi_idx*81 + lo_idx) with zero padding ----
    for (int i = tid; i < 96 * 84; i += 64) {
        int r = i / 84, c = i - r * 84;
        float v = 0.0f;
        if (r < 81 && c < 81) v = kern[r * 81 + c];
        (&K2s[0][0])[i] = v;
    }
    __syncthreads();

    const int tile = blockIdx.x * WAVES_PER_BLOCK + wv;
    const int b    = tile * 16 + m16;

    // ---- per-row 1D hat weights for dims 4..7 (w_k = max(0, 1 - |clip(x)-k|)) ----
    v4f xl = *(const v4f*)(x + b * 8 + 4);      // 16B-aligned (offset 16 within 32B row)
    if (lane < 16) {
        #pragma unroll
        for (int d = 0; d < 4; ++d) {
            float xc = fminf(fmaxf(xl[d], 0.0f), 2.0f);
            #pragma unroll
            for (int k = 0; k < 3; ++k)
                W1[wv][m16][d * 3 + k] = fmaxf(0.0f, 1.0f - fabsf(xc - (float)k));
        }
    }
    __syncthreads();

    // ---- densify V tile: V[m][lo] = prod of 4 hat weights, lo = d4*27+d5*9+d6*3+d7 ----
    for (int m = 0; m < 16; ++m) {
        for (int c = lane; c < 84; c += 32) {
            float v = 0.0f;
            if (c < 81) {
                int d4  = c / 27;
                int r27 = c - d4 * 27;
                int d5  = r27 / 9;
                int r9  = r27 - d5 * 9;
                int d6  = r9 / 3;
                int d7  = r9 - d6 * 3;
                v = W1[wv][m][d4] * W1[wv][m][3 + d5] * W1[wv][m][6 + d6] * W1[wv][m][9 + d7];
            }
            Vs[wv][m][c] = v;
        }
    }
    __syncthreads();

    // ---- GEMM: Z(16x96) = V(16x84) * K2s^T via f32 WMMA 16x16x4 ----
    // A frag: lane l holds A[m=l%16, k=2*hi+{0,1}] in 2 VGPRs  -> one ds_load_b64
    // B frag: lane l holds B[k=2*hi+{0,1}, n=l%16]             -> one ds_load_b64
    v8f acc[6] = {};
    for (int kb = 0; kb < 21; ++kb) {
        const int kcol = kb * 4 + 2 * hi;                     // even -> 8B aligned
        v2f afrag = *(const v2f*)&Vs[wv][m16][kcol];
        #pragma unroll
        for (int nt = 0; nt < 6; ++nt) {
            v2f bfrag = *(const v2f*)&K2s[nt * 16 + m16][kcol];
            acc[nt] = __builtin_amdgcn_wmma_f32_16x16x4_f32(
                false, afrag, false, bfrag, (short)0, acc[nt], false, false);
        }
    }

    // ---- spill Z accumulators: acc[nt][r] = Z[row = r + 8*hi][n = nt*16 + m16] ----
    #pragma unroll
    for (int nt = 0; nt < 6; ++nt)
        #pragma unroll
        for (int r = 0; r < 8; ++r)
            Zs[wv][r + 8 * hi][nt * 16 + m16] = acc[nt][r];
    __syncthreads();

    // ---- epilogue: out[b] = sum over 16 hi-corners of weight * Z[b, base+off] ----
    if (lane < 16) {
        v4f xh = *(const v4f*)(x + b * 8);       // dims 0..3, 32B aligned
        float t[4];
        int base = 0;
        const int str[4] = {27, 9, 3, 1};
        #pragma unroll
        for (int d = 0; d < 4; ++d) {
            float xc = fminf(fmaxf(xh[d], 0.0f), 2.0f);
            int i0 = (int)xc;
            if (i0 > 1) i0 = 1;
            t[d] = xc - (float)i0;
            base += i0 * str[d];
        }
        float s = 0.0f;
        #pragma unroll
        for (int c = 0; c < 16; ++c) {
            float w = ((c & 8) ? t[0] : 1.0f - t[0])
                    * ((c & 4) ? t[1] : 1.0f - t[1])
                    * ((c & 2) ? t[2] : 1.0f - t[2])
                    * ((c & 1) ? t[3] : 1.0f - t[3]);
            int off = ((c >> 3) & 1) * 27 + ((c >> 2) & 1) * 9 + ((c >> 1) & 1) * 3 + (c & 1);
            s += w * Zs[wv][m16][base + off];
        }
        out[b] = s;
    }
}

extern "C" void kernel_launch(void* const* d_in, const int* in_sizes, int n_in,
                              void* d_out, int out_size, void* d_ws, size_t ws_size,
                              hipStream_t stream) {
    const float* x    = (const float*)d_in[0];   // [32768*8]
    const float* kern = (const float*)d_in[1];   // [6561]
    float* out        = (float*)d_out;           // [32768]
    (void)in_sizes; (void)n_in; (void)out_size; (void)d_ws; (void)ws_size;
    Lattice_46239617908982_kernel<<<BLOCKS, 64, 0, stream>>>(x, kern, out);
}